// Transition_687194767474
// MI455X (gfx1250) — compile-verified
//
#include <hip/hip_runtime.h>
#include <hip/hip_bf16.h>

// ---------------------------------------------------------------------------
// PointNet++ set-abstraction block for MI455X (gfx1250, wave32, WMMA).
// fp32 end-to-end using V_WMMA_F32_16X16X4_F32 (reference is fp32; problem is
// L2-resident and compute-tiny, so no precision downgrade is warranted).
// BN bias terms b0/b1/b2 cancel under training-mode batch statistics.
// ---------------------------------------------------------------------------

typedef float v2f __attribute__((ext_vector_type(2)));
typedef float v8f __attribute__((ext_vector_type(8)));

#define BATCH 8
#define NPTS  4096
#define KNN   32
#define DPTS  64
#define C0IN  67
#define C0PAD 68
#define C1    64
#define C2    128
#define ROWS  64   // points per block: 2 queries * 32 neighbors

// ---- 16x16 output tile, K-dim stepped by 4, fp32 WMMA --------------------
// A (16x4 f32): lanes 0-15 hold M=lane, VGPR0=K0,VGPR1=K1; lanes 16-31 K=2,3.
// B (4x16 f32): mirrored; C/D (16x16 f32): VGPR r -> M=r (+8 for lanes>=16).
template <int CIN, int CPAD, int XS, int WS>
__device__ __forceinline__ v8f mm16x16(const float* __restrict__ Xs,
                                       const float* __restrict__ Wg,
                                       int m0, int n0, v8f acc) {
  const int lane = threadIdx.x & 31;
  const int row  = m0 + (lane & 15);
  const int col  = n0 + (lane & 15);
  const int kb   = (lane >> 4) << 1;
#pragma unroll
  for (int k = 0; k < CPAD; k += 4) {
    const int ka = k + kb;
    v2f a;
    a.x = Xs[row * XS + ka];
    a.y = Xs[row * XS + ka + 1];
    // Clamp weight K-index for the padded tail column (A is 0 there, so the
    // product contributes nothing, but the read must stay in-bounds).
    const int wk0 = (ka     < CIN) ? ka     : 0;
    const int wk1 = (ka + 1 < CIN) ? ka + 1 : 0;
    v2f b;
    b.x = Wg[col * WS + wk0];
    b.y = Wg[col * WS + wk1];
    acc = __builtin_amdgcn_wmma_f32_16x16x4_f32(false, a, false, b,
                                                (short)0, acc, false, false);
  }
  return acc;
}

// ---------------------------------------------------------------------------
// kNN: one thread per query point; candidates staged through LDS; top-32 kept
// in registers with fully-unrolled replace-max (set order is irrelevant since
// the network max-pools over neighbors).
// ---------------------------------------------------------------------------
__global__ void __launch_bounds__(256) knn_kernel(const float* __restrict__ xyz,
                                                  int* __restrict__ idxout) {
  __shared__ float cx[256], cy[256], cz[256];
  const int q = blockIdx.x * 256 + threadIdx.x;  // global query (b*N+n)
  const int b = q >> 12;
  const float qx = xyz[q * 3 + 0], qy = xyz[q * 3 + 1], qz = xyz[q * 3 + 2];

  float bestd[KNN];
  int   besti[KNN];
#pragma unroll
  for (int u = 0; u < KNN; ++u) { bestd[u] = 3.0e38f; besti[u] = 0; }
  float maxv = 3.0e38f;
  int   maxp = 0;

  for (int t = 0; t < NPTS / 256; ++t) {
    const int c = t * 256 + threadIdx.x;
    const int g = (b * NPTS + c) * 3;
    cx[threadIdx.x] = xyz[g + 0];
    cy[threadIdx.x] = xyz[g + 1];
    cz[threadIdx.x] = xyz[g + 2];
    __syncthreads();
    for (int j = 0; j < 256; ++j) {
      const float dx = cx[j] - qx, dy = cy[j] - qy, dz = cz[j] - qz;
      const float d  = fmaf(dx, dx, fmaf(dy, dy, dz * dz));
      if (d < maxv) {
        const int cand = t * 256 + j;
#pragma unroll
        for (int u = 0; u < KNN; ++u)
          if (u == maxp) { bestd[u] = d; besti[u] = cand; }
        maxv = -1.0f;
#pragma unroll
        for (int u = 0; u < KNN; ++u)
          if (bestd[u] > maxv) { maxv = bestd[u]; maxp = u; }
      }
    }
    __syncthreads();
  }
#pragma unroll
  for (int u = 0; u < KNN; ++u) idxout[q * KNN + u] = besti[u];
}

// ---------------------------------------------------------------------------
// Fused gather + MLP chain. DEPTH = 0/1/2: run chain up through layer DEPTH's
// matmul and accumulate per-channel sum/sumsq (BN batch stats). DEPTH = 3:
// full chain + ReLU + neighbor max-pool -> output.
// Block = 256 threads (8 waves) handling 2 queries (64 points).
// ---------------------------------------------------------------------------
template <int DEPTH>
__global__ void __launch_bounds__(256) fused_pass(
    const float* __restrict__ xyz, const float* __restrict__ points,
    const int* __restrict__ knn,
    const float* __restrict__ W0, const float* __restrict__ W1,
    const float* __restrict__ W2,
    const float* __restrict__ ss0, const float* __restrict__ ss1,
    const float* __restrict__ ss2,
    float* __restrict__ gstats, float* __restrict__ out) {
  __shared__ float sX[ROWS * C0PAD];  // layer input; overlaid [64x64] later
  __shared__ float sY[ROWS * C2];     // final-layer activations
  __shared__ float sStat[2 * C2];

  const int tid  = threadIdx.x;
  const int lane = tid & 31;
  const int wave = tid >> 5;
  const int q0   = blockIdx.x * 2;  // global query index (b*N+n)
  const int b    = q0 >> 12;

  if (tid < 2 * C2) sStat[tid] = 0.0f;

  // ---- gather layer-0 input: [xyz_nb - xyz_center (3) | point feats (64) | pad]
  for (int e = tid; e < ROWS * C0PAD; e += 256) {
    const int r  = e / C0PAD;
    const int c  = e - r * C0PAD;
    const int q  = q0 + (r >> 5);
    const int nb = knn[q * KNN + (r & 31)];
    float v;
    if (c < 3)          v = xyz[(b * NPTS + nb) * 3 + c] - xyz[q * 3 + c];
    else if (c < C0IN)  v = points[(b * NPTS + nb) * DPTS + (c - 3)];
    else                v = 0.0f;
    sX[e] = v;
  }
  __syncthreads();

  // ---- layer 0: [64 x 68] x W0[64 x 67]^T -> 16 tiles, 2 per wave
  v8f acc0[2];
#pragma unroll
  for (int i = 0; i < 2; ++i) {
    const int t = wave + i * 8;
    v8f z = {};
    acc0[i] = mm16x16<C0IN, C0PAD, C0PAD, C0IN>(sX, W0, (t & 3) * 16,
                                                (t >> 2) * 16, z);
  }
  if constexpr (DEPTH == 0) {
#pragma unroll
    for (int i = 0; i < 2; ++i) {
      const int t  = wave + i * 8;
      const int ch = (t >> 2) * 16 + (lane & 15);
      float s = 0.0f, s2 = 0.0f;
#pragma unroll
      for (int r = 0; r < 8; ++r) { const float v = acc0[i][r]; s += v; s2 += v * v; }
      atomicAdd(&sStat[ch], s);
      atomicAdd(&sStat[C1 + ch], s2);
    }
    __syncthreads();
    if (tid < 2 * C1) atomicAdd(&gstats[tid], sStat[tid]);
    return;
  } else {
    __syncthreads();  // all reads of sX complete before overlay writes
#pragma unroll
    for (int i = 0; i < 2; ++i) {
      const int t  = wave + i * 8;
      const int m0 = (t & 3) * 16;
      const int ch = (t >> 2) * 16 + (lane & 15);
      const float sc = ss0[ch], sh = ss0[C1 + ch];
#pragma unroll
      for (int r = 0; r < 8; ++r) {
        const int m = m0 + r + ((lane >> 4) << 3);
        sX[m * C1 + ch] = fmaxf(fmaf(acc0[i][r], sc, sh), 0.0f);
      }
    }
    __syncthreads();

    // ---- layer 1: [64 x 64] x W1[64 x 64]^T
    v8f acc1[2];
#pragma unroll
    for (int i = 0; i < 2; ++i) {
      const int t = wave + i * 8;
      v8f z = {};
      acc1[i] = mm16x16<C1, C1, C1, C1>(sX, W1, (t & 3) * 16, (t >> 2) * 16, z);
    }
    if constexpr (DEPTH == 1) {
#pragma unroll
      for (int i = 0; i < 2; ++i) {
        const int t  = wave + i * 8;
        const int ch = (t >> 2) * 16 + (lane & 15);
        float s = 0.0f, s2 = 0.0f;
#pragma unroll
        for (int r = 0; r < 8; ++r) { const float v = acc1[i][r]; s += v; s2 += v * v; }
        atomicAdd(&sStat[ch], s);
        atomicAdd(&sStat[C1 + ch], s2);
      }
      __syncthreads();
      if (tid < 2 * C1) atomicAdd(&gstats[tid], sStat[tid]);
      return;
    } else {
      __syncthreads();
#pragma unroll
      for (int i = 0; i < 2; ++i) {
        const int t  = wave + i * 8;
        const int m0 = (t & 3) * 16;
        const int ch = (t >> 2) * 16 + (lane & 15);
        const float sc = ss1[ch], sh = ss1[C1 + ch];
#pragma unroll
        for (int r = 0; r < 8; ++r) {
          const int m = m0 + r + ((lane >> 4) << 3);
          sX[m * C1 + ch] = fmaxf(fmaf(acc1[i][r], sc, sh), 0.0f);
        }
      }
      __syncthreads();

      // ---- layer 2: [64 x 64] x W2[128 x 64]^T -> 32 tiles, 4 per wave
      v8f acc2[4];
#pragma unroll
      for (int i = 0; i < 4; ++i) {
        const int t = wave + i * 8;
        v8f z = {};
        acc2[i] = mm16x16<C1, C1, C1, C1>(sX, W2, (t & 3) * 16, (t >> 2) * 16, z);
      }
      if constexpr (DEPTH == 2) {
#pragma unroll
        for (int i = 0; i < 4; ++i) {
          const int t  = wave + i * 8;
          const int ch = (t >> 2) * 16 + (lane & 15);
          float s = 0.0f, s2 = 0.0f;
#pragma unroll
          for (int r = 0; r < 8; ++r) { const float v = acc2[i][r]; s += v; s2 += v * v; }
          atomicAdd(&sStat[ch], s);
          atomicAdd(&sStat[C2 + ch], s2);
        }
        __syncthreads();
        if (tid < 2 * C2) atomicAdd(&gstats[tid], sStat[tid]);
        return;
      } else {
#pragma unroll
        for (int i = 0; i < 4; ++i) {
          const int t  = wave + i * 8;
          const int m0 = (t & 3) * 16;
          const int ch = (t >> 2) * 16 + (lane & 15);
          const float sc = ss2[ch], sh = ss2[C2 + ch];
#pragma unroll
          for (int r = 0; r < 8; ++r) {
            const int m = m0 + r + ((lane >> 4) << 3);
            sY[m * C2 + ch] = fmaxf(fmaf(acc2[i][r], sc, sh), 0.0f);
          }
        }
        __syncthreads();
        // ---- max-pool over the 32 neighbors: 256 threads = 2 queries x 128 ch
        const int l  = tid >> 7;
        const int ch = tid & 127;
        float mv = sY[(l * KNN) * C2 + ch];
#pragma unroll
        for (int r = 1; r < KNN; ++r) mv = fmaxf(mv, sY[(l * KNN + r) * C2 + ch]);
        out[(size_t)(q0 + l) * C2 + ch] = mv;
      }
    }
  }
}

// ---- tiny helpers ---------------------------------------------------------
__global__ void zero_f32(float* p, int n) {
  const int i = blockIdx.x * blockDim.x + threadIdx.x;
  if (i < n) p[i] = 0.0f;
}

__global__ void finalize_stats(const float* __restrict__ stats,
                               const float* __restrict__ g,
                               const float* __restrict__ beta,
                               float* __restrict__ ss, int C, float invM) {
  const int c = threadIdx.x;
  if (c < C) {
    const float mean = stats[c] * invM;
    const float var  = stats[C + c] * invM - mean * mean;
    const float rs   = rsqrtf(var + 1e-5f);
    const float sc   = g[c] * rs;
    ss[c]     = sc;
    ss[C + c] = beta[c] - mean * sc;  // bias b folds out under batch-stat BN
  }
}

// ---------------------------------------------------------------------------
extern "C" void kernel_launch(void* const* d_in, const int* in_sizes, int n_in,
                              void* d_out, int out_size, void* d_ws, size_t ws_size,
                              hipStream_t stream) {
  const float* xyz    = (const float*)d_in[0];
  const float* points = (const float*)d_in[1];
  const float* W0  = (const float*)d_in[2];
  // d_in[3] = b0 (cancels in training-mode BN)
  const float* g0  = (const float*)d_in[4];
  const float* be0 = (const float*)d_in[5];
  const float* W1  = (const float*)d_in[6];
  const float* g1  = (const float*)d_in[8];
  const float* be1 = (const float*)d_in[9];
  const float* W2  = (const float*)d_in[10];
  const float* g2  = (const float*)d_in[12];
  const float* be2 = (const float*)d_in[13];
  float* out = (float*)d_out;

  char* ws = (char*)d_ws;
  int* knn = (int*)ws;  // B*N*K ints = 4 MiB
  float* stats0 = (float*)(ws + (size_t)BATCH * NPTS * KNN * sizeof(int));
  float* stats1 = stats0 + 2 * C1;
  float* stats2 = stats1 + 2 * C1;
  float* ss0    = stats2 + 2 * C2;
  float* ss1    = ss0 + 2 * C1;
  float* ss2    = ss1 + 2 * C1;

  zero_f32<<<1, 1024, 0, stream>>>(stats0, 2 * (C1 + C1 + C2) + 2 * (C1 + C1 + C2));

  knn_kernel<<<BATCH * NPTS / 256, 256, 0, stream>>>(xyz, knn);

  const int   grid = BATCH * NPTS / 2;  // 2 queries per block
  const float invM = 1.0f / (float)((size_t)BATCH * NPTS * KNN);

  fused_pass<0><<<grid, 256, 0, stream>>>(xyz, points, knn, W0, W1, W2,
                                          ss0, ss1, ss2, stats0, nullptr);
  finalize_stats<<<1, C1, 0, stream>>>(stats0, g0, be0, ss0, C1, invM);

  fused_pass<1><<<grid, 256, 0, stream>>>(xyz, points, knn, W0, W1, W2,
                                          ss0, ss1, ss2, stats1, nullptr);
  finalize_stats<<<1, C1, 0, stream>>>(stats1, g1, be1, ss1, C1, invM);

  fused_pass<2><<<grid, 256, 0, stream>>>(xyz, points, knn, W0, W1, W2,
                                          ss0, ss1, ss2, stats2, nullptr);
  finalize_stats<<<1, C2, 0, stream>>>(stats2, g2, be2, ss2, C2, invM);

  fused_pass<3><<<grid, 256, 0, stream>>>(xyz, points, knn, W0, W1, W2,
                                          ss0, ss1, ss2, nullptr, out);
}